// PACEwKeypointCorrectionModule_73753178407281
// MI455X (gfx1250) — compile-verified
//
#include <hip/hip_runtime.h>
#include <hip/hip_bf16.h>
#include <math.h>

// ---------------------------------------------------------------------------
// PACE w/ keypoint correction on MI455X (gfx1250, wave32).
//
// Roofline: chamfer dominates: B*m*n = 8*4096*2048 pairwise dists * ~21 evals.
// Operands are tiny (<1MB) -> entirely L2/LDS resident; HBM (23.3TB/s) is a
// non-factor. The pairwise term (sy_j - 2 x.y) is a K=4 f32 GEMM tile ->
// V_WMMA_F32_16X16X4_F32 (2048 FLOP/instr). The packed model (-2y, sy) is
// DMA'd into LDS with tensor_load_to_lds (TDM) when available; argmin keeps
// 8 independent accumulators (one per D VGPR slot) to break the serial
// cmp/cndmask chain so it can dual-issue against the WMMA stream.
// ---------------------------------------------------------------------------

typedef __attribute__((ext_vector_type(2))) float v2f;
typedef __attribute__((ext_vector_type(8))) float v8f;
typedef __attribute__((ext_vector_type(4))) unsigned int v4u;
typedef __attribute__((ext_vector_type(4))) int v4i;
typedef __attribute__((ext_vector_type(8))) int v8i;

namespace {

constexpr int Bsz  = 8;     // batch
constexpr int Mpts = 4096;  // point cloud points
constexpr int Npts = 2048;  // cad model points
constexpr int NK   = 32;    // keypoints
constexpr int KB   = 5;     // shape basis
constexpr int NUM_STEPS = 20;
constexpr float LR = 0.5f;

// workspace layout (float offsets)
constexpr int OFF_CORR   = 0;                      // B*3*NK = 768
constexpr int OFF_R      = OFF_CORR   + 768;       // 72
constexpr int OFF_T      = OFF_R      + 72;        // 24
constexpr int OFF_C      = OFF_T      + 24;        // 40
constexpr int OFF_H      = OFF_C      + 40;        // 72
constexpr int OFF_U      = OFF_H      + 72;        // 72
constexpr int OFF_V      = OFF_U      + 72;        // 72
constexpr int OFF_S      = OFF_V      + 72;        // 24
constexpr int OFF_D      = OFF_S      + 24;        // 8
constexpr int OFF_YMEAN  = OFF_D      + 8;         // 24
constexpr int OFF_WS     = OFF_YMEAN  + 24;        // 1
constexpr int OFF_BMEAN  = OFF_WS     + 1;         // 15
constexpr int OFF_BC     = OFF_BMEAN  + 15;        // K*3*NK = 480
constexpr int OFF_GINV   = OFF_BC     + 480;       // 25
constexpr int OFF_GR     = OFF_GINV   + 25;        // 72
constexpr int OFF_GT     = OFF_GR     + 72;        // 24
constexpr int OFF_GC     = OFF_GT     + 24;        // 40
constexpr int OFF_MODEL  = 2048;                   // packed float4: B*Npts*4 = 65536
constexpr int OFF_GMODEL = OFF_MODEL  + Bsz*Npts*4;// planar: B*3*Npts = 49152

} // namespace

__device__ __forceinline__ float wred32(float v) {
#pragma unroll
  for (int o = 16; o > 0; o >>= 1) v += __shfl_xor(v, o, 32);
  return v;
}

__device__ __forceinline__ float det3(const float* m) {
  return m[0]*(m[4]*m[8]-m[5]*m[7]) - m[1]*(m[3]*m[8]-m[5]*m[6])
       + m[2]*(m[3]*m[7]-m[4]*m[6]);
}

__device__ __forceinline__ void swapcol3(float* V, int a, int b) {
#pragma unroll
  for (int r = 0; r < 3; ++r) { float t = V[r*3+a]; V[r*3+a] = V[r*3+b]; V[r*3+b] = t; }
}

// 3x3 SVD: Jacobi eigendecomposition of H^T H -> V, then U = orthonormalized H*V.
__device__ void svd3(const float* Hm, float* U, float* S, float* V) {
  float A[9];
#pragma unroll
  for (int i = 0; i < 3; ++i)
#pragma unroll
    for (int j = 0; j < 3; ++j) {
      float s = 0.f;
#pragma unroll
      for (int k = 0; k < 3; ++k) s += Hm[k*3+i] * Hm[k*3+j];
      A[i*3+j] = s;
    }
#pragma unroll
  for (int i = 0; i < 9; ++i) V[i] = 0.f;
  V[0] = V[4] = V[8] = 1.f;
  const int PP[3] = {0, 0, 1}, QQ[3] = {1, 2, 2};
  for (int it = 0; it < 12; ++it) {
    int p = PP[it % 3], q = QQ[it % 3];
    float apq = A[p*3+q];
    if (fabsf(apq) > 1e-20f) {
      float theta = (A[q*3+q] - A[p*3+p]) / (2.f * apq);
      float tt = (theta >= 0.f ? 1.f : -1.f) / (fabsf(theta) + sqrtf(theta*theta + 1.f));
      float cth = 1.f / sqrtf(tt*tt + 1.f);
      float sth = tt * cth;
#pragma unroll
      for (int k = 0; k < 3; ++k) {
        float akp = A[k*3+p], akq = A[k*3+q];
        A[k*3+p] = cth*akp - sth*akq; A[k*3+q] = sth*akp + cth*akq;
      }
#pragma unroll
      for (int k = 0; k < 3; ++k) {
        float apk = A[p*3+k], aqk = A[q*3+k];
        A[p*3+k] = cth*apk - sth*aqk; A[q*3+k] = sth*apk + cth*aqk;
      }
#pragma unroll
      for (int k = 0; k < 3; ++k) {
        float vkp = V[k*3+p], vkq = V[k*3+q];
        V[k*3+p] = cth*vkp - sth*vkq; V[k*3+q] = sth*vkp + cth*vkq;
      }
    }
  }
  float l0 = A[0], l1 = A[4], l2 = A[8];
  if (l0 < l1) { float t = l0; l0 = l1; l1 = t; swapcol3(V, 0, 1); }
  if (l0 < l2) { float t = l0; l0 = l2; l2 = t; swapcol3(V, 0, 2); }
  if (l1 < l2) { float t = l1; l1 = l2; l2 = t; swapcol3(V, 1, 2); }
  S[0] = sqrtf(fmaxf(l0, 0.f)); S[1] = sqrtf(fmaxf(l1, 0.f)); S[2] = sqrtf(fmaxf(l2, 0.f));
  float hv0[3], hv1[3], u0[3], u1[3], u2[3];
#pragma unroll
  for (int i = 0; i < 3; ++i) {
    hv0[i] = Hm[i*3+0]*V[0] + Hm[i*3+1]*V[3] + Hm[i*3+2]*V[6];
    hv1[i] = Hm[i*3+0]*V[1] + Hm[i*3+1]*V[4] + Hm[i*3+2]*V[7];
  }
  float n0 = sqrtf(hv0[0]*hv0[0] + hv0[1]*hv0[1] + hv0[2]*hv0[2]);
  if (n0 > 1e-12f) { u0[0]=hv0[0]/n0; u0[1]=hv0[1]/n0; u0[2]=hv0[2]/n0; }
  else             { u0[0]=1.f; u0[1]=0.f; u0[2]=0.f; }
  float pr = u0[0]*hv1[0] + u0[1]*hv1[1] + u0[2]*hv1[2];
  float w1[3] = {hv1[0]-pr*u0[0], hv1[1]-pr*u0[1], hv1[2]-pr*u0[2]};
  float n1 = sqrtf(w1[0]*w1[0] + w1[1]*w1[1] + w1[2]*w1[2]);
  if (n1 > 1e-12f) { u1[0]=w1[0]/n1; u1[1]=w1[1]/n1; u1[2]=w1[2]/n1; }
  else {
    float ax[3] = {1.f, 0.f, 0.f};
    if (fabsf(u0[0]) > 0.9f) { ax[0]=0.f; ax[1]=1.f; }
    u1[0] = u0[1]*ax[2]-u0[2]*ax[1]; u1[1] = u0[2]*ax[0]-u0[0]*ax[2]; u1[2] = u0[0]*ax[1]-u0[1]*ax[0];
    float nn = sqrtf(u1[0]*u1[0]+u1[1]*u1[1]+u1[2]*u1[2]);
    u1[0]/=nn; u1[1]/=nn; u1[2]/=nn;
  }
  u2[0] = u0[1]*u1[2]-u0[2]*u1[1]; u2[1] = u0[2]*u1[0]-u0[0]*u1[2]; u2[2] = u0[0]*u1[1]-u0[1]*u1[0];
#pragma unroll
  for (int i = 0; i < 3; ++i) { U[i*3+0]=u0[i]; U[i*3+1]=u1[i]; U[i*3+2]=u2[i]; }
}

// ---------------------------------------------------------------------------
// Precompute: Bmean, Bc, Ginv (constant across all steps), corr init copy.
__global__ void k_precompute(const float* __restrict__ corr0,
                             const float* __restrict__ mkp,
                             const float* __restrict__ w,
                             const float* __restrict__ lamp,
                             float* __restrict__ W) {
  const int p = threadIdx.x;                 // keypoint lane
  const float lam = lamp[0];
  float wn = w[p];
  float wssum = wred32(wn);
  float bc[KB][3], bmn[KB][3];
#pragma unroll
  for (int k = 0; k < KB; ++k)
#pragma unroll
    for (int i = 0; i < 3; ++i) {
      float v = mkp[(k*3+i)*NK + p];
      float mean = wred32(wn * v) / wssum;
      bmn[k][i] = mean;
      bc[k][i] = v - mean;
      W[OFF_BC + (k*3+i)*NK + p] = bc[k][i];
    }
  float Gm[KB*KB];
#pragma unroll
  for (int k = 0; k < KB; ++k)
#pragma unroll
    for (int l = 0; l < KB; ++l) {
      float d = bc[k][0]*bc[l][0] + bc[k][1]*bc[l][1] + bc[k][2]*bc[l][2];
      float s = wred32(wn * d);
      Gm[k*KB+l] = s + (k == l ? lam : 0.f);
    }
  float a[KB][2*KB];
#pragma unroll
  for (int k = 0; k < KB; ++k)
#pragma unroll
    for (int l = 0; l < 2*KB; ++l)
      a[k][l] = (l < KB) ? Gm[k*KB+l] : (l - KB == k ? 1.f : 0.f);
  for (int col = 0; col < KB; ++col) {
    float inv = 1.f / a[col][col];
    for (int l = 0; l < 2*KB; ++l) a[col][l] *= inv;
    for (int r = 0; r < KB; ++r) {
      if (r == col) continue;
      float f = a[r][col];
      for (int l = 0; l < 2*KB; ++l) a[r][l] -= f * a[col][l];
    }
  }
  if (p == 0) {
    for (int k = 0; k < KB; ++k)
      for (int l = 0; l < KB; ++l) W[OFF_GINV + k*KB + l] = a[k][KB + l];
    for (int k = 0; k < KB; ++k)
      for (int i = 0; i < 3; ++i) W[OFF_BMEAN + k*3 + i] = bmn[k][i];
    W[OFF_WS] = wssum;
  }
  for (int idx = p; idx < Bsz*3*NK; idx += 32) W[OFF_CORR + idx] = corr0[idx];
}

// ---------------------------------------------------------------------------
// PACE forward: one wave per batch.
__global__ void k_pace_forward(const float* __restrict__ kp,
                               const float* __restrict__ w,
                               const float* __restrict__ lamp,
                               float* __restrict__ W) {
  const int b = blockIdx.x, p = threadIdx.x;
  const float lam = lamp[0];
  float wn = w[p];
  float wssum = wred32(wn);
  float y[3], ym[3], yc[3];
#pragma unroll
  for (int i = 0; i < 3; ++i) {
    y[i] = kp[(b*3+i)*NK + p] + W[OFF_CORR + (b*3+i)*NK + p];
    ym[i] = wred32(wn * y[i]) / wssum;
    yc[i] = y[i] - ym[i];
  }
  float bc[KB][3], bmn[KB][3], Gi[KB*KB];
#pragma unroll
  for (int k = 0; k < KB; ++k)
#pragma unroll
    for (int i = 0; i < 3; ++i) {
      bc[k][i] = W[OFF_BC + (k*3+i)*NK + p];
      bmn[k][i] = W[OFF_BMEAN + k*3 + i];
    }
  for (int q = 0; q < KB*KB; ++q) Gi[q] = W[OFF_GINV + q];

  float c[KB];
#pragma unroll
  for (int k = 0; k < KB; ++k) c[k] = 1.f / KB;

  float Hm[9], U[9], S[3], V[9], R[9], dsign = 1.f;
  for (int iter = 0; iter < 3; ++iter) {   // 2 PACE iters + final procrustes
    float bp[3];
#pragma unroll
    for (int i = 0; i < 3; ++i) {
      bp[i] = 0.f;
#pragma unroll
      for (int k = 0; k < KB; ++k) bp[i] += c[k] * bc[k][i];
    }
#pragma unroll
    for (int i = 0; i < 3; ++i)
#pragma unroll
      for (int j = 0; j < 3; ++j)
        Hm[i*3+j] = wred32(wn * yc[i] * bp[j]);
    svd3(Hm, U, S, V);
    dsign = det3(Hm) >= 0.f ? 1.f : -1.f;
#pragma unroll
    for (int i = 0; i < 3; ++i)
#pragma unroll
      for (int j = 0; j < 3; ++j) {
        float s = 0.f;
#pragma unroll
        for (int k = 0; k < 3; ++k) s += U[i*3+k] * (k == 2 ? dsign : 1.f) * V[j*3+k];
        R[i*3+j] = s;
      }
    if (iter == 2) break;
    float yr[3];
#pragma unroll
    for (int i = 0; i < 3; ++i)
      yr[i] = R[0*3+i]*yc[0] + R[1*3+i]*yc[1] + R[2*3+i]*yc[2];
    float h[KB];
#pragma unroll
    for (int k = 0; k < KB; ++k) {
      float d = bc[k][0]*yr[0] + bc[k][1]*yr[1] + bc[k][2]*yr[2];
      h[k] = wred32(wn * d) + lam / KB;
    }
#pragma unroll
    for (int k = 0; k < KB; ++k) {
      float s = 0.f;
#pragma unroll
      for (int l = 0; l < KB; ++l) s += Gi[k*KB+l] * h[l];
      c[k] = s;
    }
  }
  if (p == 0) {
    float bmean[3];
#pragma unroll
    for (int i = 0; i < 3; ++i) {
      bmean[i] = 0.f;
#pragma unroll
      for (int k = 0; k < KB; ++k) bmean[i] += c[k] * bmn[k][i];
    }
    for (int q = 0; q < 9; ++q) { W[OFF_R + b*9 + q] = R[q]; W[OFF_H + b*9 + q] = Hm[q];
                                  W[OFF_U + b*9 + q] = U[q]; W[OFF_V + b*9 + q] = V[q]; }
    for (int i = 0; i < 3; ++i) {
      W[OFF_T + b*3 + i] = ym[i] - (R[i*3+0]*bmean[0] + R[i*3+1]*bmean[1] + R[i*3+2]*bmean[2]);
      W[OFF_S + b*3 + i] = S[i];
      W[OFF_YMEAN + b*3 + i] = ym[i];
    }
    for (int k = 0; k < KB; ++k) W[OFF_C + b*KB + k] = c[k];
    W[OFF_D + b] = dsign;
  }
}

// ---------------------------------------------------------------------------
// model points, pre-packed for the WMMA A operand: (-2y0,-2y1,-2y2,sy) float4.
// Also zeroes the (planar) gModel accumulator.
__global__ void k_model(const float* __restrict__ cad, float* __restrict__ W) {
  const int j = blockIdx.x * blockDim.x + threadIdx.x;
  const int b = blockIdx.y;
  if (j >= Npts) return;
  float c[KB], R[9], t[3];
  for (int k = 0; k < KB; ++k) c[k] = W[OFF_C + b*KB + k];
  for (int q = 0; q < 9; ++q) R[q] = W[OFF_R + b*9 + q];
  for (int i = 0; i < 3; ++i) t[i] = W[OFF_T + b*3 + i];
  float Mv[3], y[3];
#pragma unroll
  for (int i = 0; i < 3; ++i) {
    float s = 0.f;
#pragma unroll
    for (int q = 0; q < KB; ++q) s += c[q] * cad[(q*3+i)*Npts + j];
    Mv[i] = s;
  }
#pragma unroll
  for (int i = 0; i < 3; ++i) {
    y[i] = R[i*3+0]*Mv[0] + R[i*3+1]*Mv[1] + R[i*3+2]*Mv[2] + t[i];
    W[OFF_GMODEL + (b*3+i)*Npts + j] = 0.f;
  }
  float4* pk = (float4*)(W + OFF_MODEL);
  pk[b*Npts + j] = make_float4(-2.f*y[0], -2.f*y[1], -2.f*y[2],
                               y[0]*y[0] + y[1]*y[1] + y[2]*y[2]);
}

// ---------------------------------------------------------------------------
// Chamfer argmin via V_WMMA_F32_16X16X4_F32 + gradient scatter into gModel.
// Block = 256 threads = 8 waves; each wave owns a 16-wide tile of pc indices;
// packed model (32KB, contiguous) is DMA'd into LDS with the TDM when the
// toolchain exposes tensor_load_to_lds.
__global__ void __launch_bounds__(256)
k_chamfer(const float* __restrict__ pc, float* __restrict__ W) {
  __shared__ float4 smodel[Npts];            // ONLY shared object -> LDS offset 0
  const int b = blockIdx.y;
  const int tid = threadIdx.x;
  const int wave = tid >> 5, lane = tid & 31, l16 = lane & 31 & 15;
  const int i = blockIdx.x * 128 + wave * 16 + (lane & 15);  // pc point this lane owns

  // prefetch pc operands (global_prefetch_b8) while the model is staged
  __builtin_prefetch(&pc[(b*3+0)*Mpts + i], 0, 1);
  __builtin_prefetch(&pc[(b*3+2)*Mpts + i], 0, 1);

#if __has_builtin(__builtin_amdgcn_tensor_load_to_lds)
  if (wave == 0) {
    // Tensor DMA descriptor: 2D tensor of 4096 x 8B words (= Npts float4s),
    // tile == whole tensor, destination LDS offset 0.
    unsigned long long ga =
        (unsigned long long)(uintptr_t)(W + OFF_MODEL) +
        (unsigned long long)b * (unsigned long long)(Npts * 16);
    v4u g0;
    g0.x = 1u;                                        // count=1 valid descriptor
    g0.y = 0u;                                        // lds_addr = 0
    g0.z = (unsigned)(ga & 0xFFFFFFFFu);              // global_addr[31:0]
    g0.w = (unsigned)((ga >> 32) & 0x01FFFFFFu) | (2u << 30);  // addr[56:32] | type=2
    v8i g1 = {
      0x30000,                 // data_size = 3 (8 bytes)
      (int)(4096u << 16),      // tensor_dim0[15:0] = 4096 (8B units)
      0x10000,                 // tensor_dim1 = 1
      (int)(4096u << 16),      // tile_dim0 = 4096
      1,                       // tile_dim1 = 1
      4096,                    // tensor_dim0_stride[31:0] = 4096
      (int)(4096u << 16),      // tensor_dim1_stride[15:0] = 4096
      0
    };
    v4i gz = {0, 0, 0, 0};
#if __clang_major__ >= 23
    v8i gz8 = {0, 0, 0, 0, 0, 0, 0, 0};
    __builtin_amdgcn_tensor_load_to_lds(g0, g1, gz, gz, gz8, 0);
#else
    __builtin_amdgcn_tensor_load_to_lds(g0, g1, gz, gz, 0);
#endif
    __builtin_amdgcn_s_wait_tensorcnt(0);
  }
#else
  {
    const float4* pk = (const float4*)(W + OFF_MODEL) + b * Npts;
    for (int j = tid; j < Npts; j += 256) smodel[j] = pk[j];
  }
#endif
  // The TDM writes LDS behind the compiler's back: escape smodel's address
  // into an opaque asm with a memory clobber so the smodel loads below are
  // not folded away as loads of never-written memory.
  asm volatile("" : : "v"((const void*)&smodel[0]) : "memory");
  __syncthreads();

  const float x0 = pc[(b*3+0)*Mpts + i];
  const float x1 = pc[(b*3+1)*Mpts + i];
  const float x2 = pc[(b*3+2)*Mpts + i];
  // B operand (4x16, K x N): lanes 0-15 hold K=0/1, lanes 16-31 hold K=2/3
  v2f bb;
  bb.x = (lane < 16) ? x0 : x2;
  bb.y = (lane < 16) ? x1 : 1.0f;

  // 8 independent argmin accumulators (one per D VGPR slot) -> dependency
  // distance of a full tile per chain, so cmp/cndmask dual-issue cleanly.
  float minv[8];
  int   minj[8];
#pragma unroll
  for (int v = 0; v < 8; ++v) { minv[v] = 3.4e38f; minj[v] = 0; }

  const int off8 = (lane >> 4) << 3;   // M rows 0-7 or 8-15 of each tile
  for (int jt = 0; jt < Npts / 16; ++jt) {
    float4 va = smodel[jt*16 + l16];
    // A operand (16x4, M x K): lanes 0-15 hold K=0/1, lanes 16-31 hold K=2/3
    v2f a;
    a.x = (lane < 16) ? va.x : va.z;
    a.y = (lane < 16) ? va.y : va.w;
    v8f cz = {0.f, 0.f, 0.f, 0.f, 0.f, 0.f, 0.f, 0.f};
    union { v8f v; float f[8]; } d;
    // D[j][i] = sy_j - 2 x_i . y_j  (sx_i constant per row: argmin-irrelevant)
    d.v = __builtin_amdgcn_wmma_f32_16x16x4_f32(false, a, false, bb,
                                                (short)0, cz, false, false);
    const int mbase = jt*16 + off8;
#pragma unroll
    for (int v = 0; v < 8; ++v) {
      float val = d.f[v];
      if (val < minv[v]) { minv[v] = val; minj[v] = mbase + v; }
    }
  }
  // reduce the 8 slot accumulators (branchless: two cndmasks per step)
  float mv = minv[0];
  int   mj = minj[0];
#pragma unroll
  for (int v = 1; v < 8; ++v) {
    bool better = (minv[v] < mv) | ((minv[v] == mv) & (minj[v] < mj));
    mv = better ? minv[v] : mv;
    mj = better ? minj[v] : mj;
  }
  // combine the two half-lane row groups (M 0-7 vs 8-15 per tile)
  float ov = __shfl_xor(mv, 16, 32);
  int   oj = __shfl_xor(mj, 16, 32);
  {
    bool better = (ov < mv) | ((ov == mv) & (oj < mj));
    mv = better ? ov : mv;
    mj = better ? oj : mj;
  }

  if (lane < 16) {
    // dL/dY_{j*} += (2/m) (Y_{j*} - X_i)
    float4 vm = smodel[mj];
    const float gs = 2.0f / (float)Mpts;
    float yj[3] = {-0.5f*vm.x, -0.5f*vm.y, -0.5f*vm.z};
    float xi[3] = {x0, x1, x2};
#pragma unroll
    for (int k = 0; k < 3; ++k)
      atomicAdd(&W[OFF_GMODEL + (b*3+k)*Npts + mj], gs * (yj[k] - xi[k]));
  }
}

// ---------------------------------------------------------------------------
// reduce gModel -> gt(3), gR(9), gc(5) per batch
__global__ void __launch_bounds__(256)
k_reduce(const float* __restrict__ cad, float* __restrict__ W) {
  __shared__ float sred[256];
  const int b = blockIdx.x, tid = threadIdx.x;
  float c[KB], R[9];
  for (int k = 0; k < KB; ++k) c[k] = W[OFF_C + b*KB + k];
  for (int q = 0; q < 9; ++q) R[q] = W[OFF_R + b*9 + q];
  float agt[3] = {0,0,0}, agR[9] = {0,0,0,0,0,0,0,0,0}, agc[KB] = {0,0,0,0,0};
  for (int j = tid; j < Npts; j += 256) {
    float g[3], Mv[3];
#pragma unroll
    for (int l = 0; l < 3; ++l) g[l] = W[OFF_GMODEL + (b*3+l)*Npts + j];
#pragma unroll
    for (int l = 0; l < 3; ++l) {
      float s = 0.f;
#pragma unroll
      for (int q = 0; q < KB; ++q) s += c[q] * cad[(q*3+l)*Npts + j];
      Mv[l] = s;
      agt[l] += g[l];
    }
#pragma unroll
    for (int ii = 0; ii < 3; ++ii)
#pragma unroll
      for (int l = 0; l < 3; ++l) agR[ii*3+l] += g[ii] * Mv[l];
    float rtg[3];
#pragma unroll
    for (int ii = 0; ii < 3; ++ii)
      rtg[ii] = R[0*3+ii]*g[0] + R[1*3+ii]*g[1] + R[2*3+ii]*g[2];
#pragma unroll
    for (int q = 0; q < KB; ++q) {
      float s = 0.f;
#pragma unroll
      for (int ii = 0; ii < 3; ++ii) s += rtg[ii] * cad[(q*3+ii)*Npts + j];
      agc[q] += s;
    }
  }
  float acc[17];
  for (int l = 0; l < 3; ++l) acc[l] = agt[l];
  for (int q = 0; q < 9; ++q) acc[3+q] = agR[q];
  for (int q = 0; q < KB; ++q) acc[12+q] = agc[q];
  for (int v = 0; v < 17; ++v) {
    sred[tid] = acc[v];
    __syncthreads();
    for (int s = 128; s > 0; s >>= 1) {
      if (tid < s) sred[tid] += sred[tid + s];
      __syncthreads();
    }
    if (tid == 0) {
      float r = sred[0];
      if (v < 3)       W[OFF_GT + b*3 + v] = r;
      else if (v < 12) W[OFF_GR + b*9 + (v-3)] = r;
      else             W[OFF_GC + b*KB + (v-12)] = r;
    }
    __syncthreads();
  }
}

// ---------------------------------------------------------------------------
// backward through PACE (exact t/ymean/h paths + analytic SVD backward for R,
// truncated at the final iteration), then corr -= LR * gcorr.
__global__ void k_pace_backward(const float* __restrict__ kp,
                                const float* __restrict__ w,
                                float* __restrict__ W) {
  const int b = blockIdx.x, p = threadIdx.x;
  float wn = w[p];
  float wssum = W[OFF_WS];
  float gR[9], gt[3], gc[KB], U[9], V[9], S[3], R[9], c[KB], ym[3];
  for (int q = 0; q < 9; ++q) { gR[q] = W[OFF_GR + b*9 + q]; U[q] = W[OFF_U + b*9 + q];
                                V[q] = W[OFF_V + b*9 + q];  R[q] = W[OFF_R + b*9 + q]; }
  for (int i = 0; i < 3; ++i) { gt[i] = W[OFF_GT + b*3 + i]; S[i] = W[OFF_S + b*3 + i];
                                ym[i] = W[OFF_YMEAN + b*3 + i]; }
  for (int k = 0; k < KB; ++k) { gc[k] = W[OFF_GC + b*KB + k]; c[k] = W[OFF_C + b*KB + k]; }
  const float dsg = W[OFF_D + b];

  float bc[KB][3], bmn[KB][3], Gi[KB*KB];
#pragma unroll
  for (int k = 0; k < KB; ++k)
#pragma unroll
    for (int i = 0; i < 3; ++i) {
      bc[k][i] = W[OFF_BC + (k*3+i)*NK + p];
      bmn[k][i] = W[OFF_BMEAN + k*3 + i];
    }
  for (int q = 0; q < KB*KB; ++q) Gi[q] = W[OFF_GINV + q];

  // step 1: t = ymean - R bmean(c)
  float bmean[3];
#pragma unroll
  for (int i = 0; i < 3; ++i) {
    bmean[i] = 0.f;
#pragma unroll
    for (int k = 0; k < KB; ++k) bmean[i] += c[k] * bmn[k][i];
  }
  float gymean[3] = {gt[0], gt[1], gt[2]};
#pragma unroll
  for (int i = 0; i < 3; ++i)
#pragma unroll
    for (int j = 0; j < 3; ++j) gR[i*3+j] -= gt[i] * bmean[j];
#pragma unroll
  for (int k = 0; k < KB; ++k) {
    float rb0 = R[0]*bmn[k][0] + R[1]*bmn[k][1] + R[2]*bmn[k][2];
    float rb1 = R[3]*bmn[k][0] + R[4]*bmn[k][1] + R[5]*bmn[k][2];
    float rb2 = R[6]*bmn[k][0] + R[7]*bmn[k][1] + R[8]*bmn[k][2];
    gc[k] -= gt[0]*rb0 + gt[1]*rb1 + gt[2]*rb2;
  }

  // step 2: SVD backward, R = U diag(1,1,d) V^T (d piecewise-const)
  float Dg[3] = {1.f, 1.f, dsg};
  float gU[9], gV[9];
#pragma unroll
  for (int i = 0; i < 3; ++i)
#pragma unroll
    for (int k = 0; k < 3; ++k) {
      float su = 0.f, sv = 0.f;
#pragma unroll
      for (int j = 0; j < 3; ++j) { su += gR[i*3+j] * V[j*3+k]; sv += gR[j*3+i] * U[j*3+k]; }
      gU[i*3+k] = Dg[k] * su;
      gV[i*3+k] = Dg[k] * sv;
    }
  float Ah[9], Bh[9];
#pragma unroll
  for (int i = 0; i < 3; ++i)
#pragma unroll
    for (int j = 0; j < 3; ++j) {
      float sa = 0.f, sb = 0.f;
#pragma unroll
      for (int k = 0; k < 3; ++k) { sa += U[k*3+i] * gU[k*3+j]; sb += V[k*3+i] * gV[k*3+j]; }
      Ah[i*3+j] = sa; Bh[i*3+j] = sb;
    }
  float Tm[9];
#pragma unroll
  for (int i = 0; i < 3; ++i)
#pragma unroll
    for (int j = 0; j < 3; ++j) {
      if (i == j) { Tm[i*3+j] = 0.f; continue; }
      float den = S[j]*S[j] - S[i]*S[i];
      float F = den / (den*den + 1e-6f);
      Tm[i*3+j] = F * ((Ah[i*3+j] - Ah[j*3+i]) * S[j] + S[i] * (Bh[i*3+j] - Bh[j*3+i]));
    }
  float gH[9];
#pragma unroll
  for (int i = 0; i < 3; ++i)
#pragma unroll
    for (int j = 0; j < 3; ++j) {
      float s = 0.f;
#pragma unroll
      for (int a = 0; a < 3; ++a)
#pragma unroll
        for (int e = 0; e < 3; ++e) s += U[i*3+a] * Tm[a*3+e] * V[j*3+e];
      gH[i*3+j] = s;
    }

  // step 3: H = sum_p wn yc_p b_p^T
  float yc[3], bp[3];
#pragma unroll
  for (int i = 0; i < 3; ++i) {
    float y = kp[(b*3+i)*NK + p] + W[OFF_CORR + (b*3+i)*NK + p];
    yc[i] = y - ym[i];
    bp[i] = 0.f;
#pragma unroll
    for (int k = 0; k < KB; ++k) bp[i] += c[k] * bc[k][i];
  }
  float gyc[3], gbp[3];
#pragma unroll
  for (int i = 0; i < 3; ++i) {
    gyc[i] = wn * (gH[i*3+0]*bp[0] + gH[i*3+1]*bp[1] + gH[i*3+2]*bp[2]);
    gbp[i] = wn * (gH[0*3+i]*yc[0] + gH[1*3+i]*yc[1] + gH[2*3+i]*yc[2]);
  }
#pragma unroll
  for (int k = 0; k < KB; ++k)
    gc[k] += wred32(gbp[0]*bc[k][0] + gbp[1]*bc[k][1] + gbp[2]*bc[k][2]);

  // step 4: c = Ginv h, h from Bc.(R^T yc)
  float gh[KB];
#pragma unroll
  for (int k = 0; k < KB; ++k) {
    float s = 0.f;
#pragma unroll
    for (int l = 0; l < KB; ++l) s += Gi[k*KB+l] * gc[l];
    gh[k] = s;
  }
  float gyr[3];
#pragma unroll
  for (int i = 0; i < 3; ++i) {
    float s = 0.f;
#pragma unroll
    for (int k = 0; k < KB; ++k) s += gh[k] * bc[k][i];
    gyr[i] = wn * s;
  }
#pragma unroll
  for (int i = 0; i < 3; ++i)
    gyc[i] += R[i*3+0]*gyr[0] + R[i*3+1]*gyr[1] + R[i*3+2]*gyr[2];

  // step 5+6: yc = y - ymean, ymean = sum w y / ws ; update corr
#pragma unroll
  for (int i = 0; i < 3; ++i) {
    float tot = wred32(gyc[i]);
    float gy = gyc[i] + (wn / wssum) * (gymean[i] - tot);
    W[OFF_CORR + (b*3+i)*NK + p] -= LR * gy;
  }
}

// ---------------------------------------------------------------------------
__global__ void k_finalize(const float* __restrict__ kp,
                           const float* __restrict__ W,
                           float* __restrict__ out) {
  const int tid = threadIdx.x;
  for (int q = tid; q < 72; q += 256) out[q] = W[OFF_R + q];          // R
  for (int q = tid; q < 24; q += 256) out[72 + q] = W[OFF_T + q];     // t
  for (int q = tid; q < 40; q += 256) out[96 + q] = W[OFF_C + q];     // c
  for (int q = tid; q < 768; q += 256) {
    float corr = W[OFF_CORR + q];
    out[136 + q] = corr;                                              // corr
    out[904 + q] = kp[q] + corr;                                      // corrected
  }
}

// ---------------------------------------------------------------------------
extern "C" void kernel_launch(void* const* d_in, const int* in_sizes, int n_in,
                              void* d_out, int out_size, void* d_ws, size_t ws_size,
                              hipStream_t stream) {
  const float* pc   = (const float*)d_in[0];
  const float* kp   = (const float*)d_in[1];
  const float* cor0 = (const float*)d_in[2];
  const float* mkp  = (const float*)d_in[3];
  const float* cad  = (const float*)d_in[4];
  const float* wgt  = (const float*)d_in[5];
  const float* lam  = (const float*)d_in[6];
  float* W = (float*)d_ws;
  float* out = (float*)d_out;
  (void)in_sizes; (void)n_in; (void)out_size; (void)ws_size;

  k_precompute<<<1, 32, 0, stream>>>(cor0, mkp, wgt, lam, W);
  for (int step = 0; step < NUM_STEPS; ++step) {
    k_pace_forward<<<Bsz, 32, 0, stream>>>(kp, wgt, lam, W);
    k_model<<<dim3(Npts / 256, Bsz), 256, 0, stream>>>(cad, W);
    k_chamfer<<<dim3(Mpts / 128, Bsz), 256, 0, stream>>>(pc, W);
    k_reduce<<<Bsz, 256, 0, stream>>>(cad, W);
    k_pace_backward<<<Bsz, 32, 0, stream>>>(kp, wgt, W);
  }
  k_pace_forward<<<Bsz, 32, 0, stream>>>(kp, wgt, lam, W);
  k_finalize<<<1, 256, 0, stream>>>(kp, W, out);
}